// EucSageencoder_9869834846905
// MI455X (gfx1250) — compile-verified
//
#include <hip/hip_runtime.h>

typedef __attribute__((ext_vector_type(2))) float v2f;
typedef __attribute__((ext_vector_type(8))) float v8f;

#define N_NODES 50000
#define N_EDGES 800000
#define IN_DIM  128
#define HID_DIM 256
#define OUT_DIM 128

// ---------------------------------------------------------------------------
// Scatter-accumulate: one wave (32 lanes) per edge.
//   agg[dst] += X[src]  (per-feature float atomics, L2-resident gathers)
//   deg[dst] += 1       (lane 0)
// ---------------------------------------------------------------------------
__global__ void __launch_bounds__(256) sage_scatter(
    const float* __restrict__ X,
    const int*   __restrict__ src,
    const int*   __restrict__ dst,
    float*       __restrict__ agg,
    float*       __restrict__ deg,
    int nEdges, int F)
{
    int gid  = blockIdx.x * blockDim.x + threadIdx.x;
    int e    = gid >> 5;
    int lane = gid & 31;
    if (e >= nEdges) return;

    int s = src[e];
    int d = dst[e];
    if (lane == 0) atomicAdd(deg + d, 1.0f);

    const float* xs = X + (size_t)s * F;
    float*       ad = agg + (size_t)d * F;
    for (int f = lane; f < F; f += 32)
        atomicAdd(ad + f, xs[f]);
}

// ---------------------------------------------------------------------------
// agg[i, :] /= max(deg[i], 1)
// ---------------------------------------------------------------------------
__global__ void __launch_bounds__(256) sage_normalize(
    float* __restrict__ agg, const float* __restrict__ deg, int n, int F)
{
    int i = blockIdx.x * blockDim.x + threadIdx.x;
    if (i >= n * F) return;
    float dv = deg[i / F];
    agg[i] *= 1.0f / fmaxf(dv, 1.0f);
}

// ---------------------------------------------------------------------------
// Fused GEMM: Out = relu([Aself | Aagg] @ W + bias)
//   Aself, Aagg : [nRows, half] each (virtual concat along K, K = 2*half)
//   W           : [2*half, nCols] row-major
// One wave computes one 16x16 tile of Out using V_WMMA_F32_16X16X4_F32.
// Fragment layouts per CDNA5 ISA 7.12.2:
//   A (16x4 f32): lanes 0-15 -> K={k,k+1}, lanes 16-31 -> K={k+2,k+3}, M=lane%16
//   B (4x16 f32): lanes 0-15 -> rows {k,k+1}, lanes 16-31 -> rows {k+2,k+3}, N=lane%16
//   C/D: VGPR j -> M = j + 8*(lane>=16), N = lane%16
// ---------------------------------------------------------------------------
__global__ void __launch_bounds__(256) sage_gemm_relu(
    const float* __restrict__ Aself,
    const float* __restrict__ Aagg,
    const float* __restrict__ W,
    const float* __restrict__ bias,
    float*       __restrict__ Out,
    int nRows, int half, int nCols)
{
    int gwave = (blockIdx.x * blockDim.x + threadIdx.x) >> 5;
    int lane  = threadIdx.x & 31;

    int colTiles = nCols >> 4;
    int rowTile  = gwave / colTiles;
    int colTile  = gwave - rowTile * colTiles;
    if (rowTile * 16 >= nRows) return;

    int row0 = rowTile << 4;
    int col0 = colTile << 4;
    int lm   = lane & 15;   // M (for A) / N (for B,C,D)
    int hi   = lane >> 4;   // 0: K pair {k,k+1}; 1: K pair {k+2,k+3}

    int arow = row0 + lm;
    int bcol = col0 + lm;

    // bias depends only on the output column -> same value in all 8 acc regs
    float bv = bias[bcol];
    v8f acc;
#pragma unroll
    for (int j = 0; j < 8; ++j) acc[j] = bv;

    const float* Arow0 = Aself + (size_t)arow * half;
    const float* Arow1 = Aagg  + (size_t)arow * half;

    // ---- first K half: self features ----
    for (int k = 0; k < half; k += 4) {
        int kk = k + (hi << 1);
        v2f a = *(const v2f*)(Arow0 + kk);          // 8B aligned (kk even)
        v2f b;
        b.x = W[(size_t)kk       * nCols + bcol];
        b.y = W[(size_t)(kk + 1) * nCols + bcol];
        acc = __builtin_amdgcn_wmma_f32_16x16x4_f32(
            false, a, false, b, (short)0, acc, false, false);
    }
    // ---- second K half: aggregated neighbor features ----
    for (int k = 0; k < half; k += 4) {
        int kk = k + (hi << 1);
        v2f a = *(const v2f*)(Arow1 + kk);
        int kg = half + kk;
        v2f b;
        b.x = W[(size_t)kg       * nCols + bcol];
        b.y = W[(size_t)(kg + 1) * nCols + bcol];
        acc = __builtin_amdgcn_wmma_f32_16x16x4_f32(
            false, a, false, b, (short)0, acc, false, false);
    }

    // ---- ReLU + store ----
#pragma unroll
    for (int j = 0; j < 8; ++j) {
        int r = row0 + j + (hi << 3);
        float v = acc[j] > 0.0f ? acc[j] : 0.0f;
        Out[(size_t)r * nCols + bcol] = v;
    }
}

// ---------------------------------------------------------------------------
// Launch
// ---------------------------------------------------------------------------
extern "C" void kernel_launch(void* const* d_in, const int* in_sizes, int n_in,
                              void* d_out, int out_size, void* d_ws, size_t ws_size,
                              hipStream_t stream)
{
    const float* x   = (const float*)d_in[0];
    const float* W1  = (const float*)d_in[1];
    const float* b1  = (const float*)d_in[2];
    const float* W2  = (const float*)d_in[3];
    const float* b2  = (const float*)d_in[4];
    const int* src0  = (const int*)d_in[5];
    const int* dst0  = (const int*)d_in[6];
    const int* src1  = (const int*)d_in[7];
    const int* dst1  = (const int*)d_in[8];
    float* out = (float*)d_out;

    // workspace: h [N,HID] | agg [N,HID max] | deg [N]
    float* h   = (float*)d_ws;
    float* agg = h   + (size_t)N_NODES * HID_DIM;
    float* deg = agg + (size_t)N_NODES * HID_DIM;

    const int BLK = 256;

    // ===== layer 1 =====
    hipMemsetAsync(agg, 0, (size_t)N_NODES * IN_DIM * sizeof(float), stream);
    hipMemsetAsync(deg, 0, (size_t)N_NODES * sizeof(float), stream);

    {
        long long thr = (long long)N_EDGES * 32;
        sage_scatter<<<(int)((thr + BLK - 1) / BLK), BLK, 0, stream>>>(
            x, src0, dst0, agg, deg, N_EDGES, IN_DIM);
    }
    {
        int total = N_NODES * IN_DIM;
        sage_normalize<<<(total + BLK - 1) / BLK, BLK, 0, stream>>>(
            agg, deg, N_NODES, IN_DIM);
    }
    {
        int waves = (N_NODES / 16) * (HID_DIM / 16);
        long long thr = (long long)waves * 32;
        sage_gemm_relu<<<(int)((thr + BLK - 1) / BLK), BLK, 0, stream>>>(
            x, agg, W1, b1, h, N_NODES, IN_DIM, HID_DIM);
    }

    // ===== layer 2 =====
    hipMemsetAsync(agg, 0, (size_t)N_NODES * HID_DIM * sizeof(float), stream);
    hipMemsetAsync(deg, 0, (size_t)N_NODES * sizeof(float), stream);

    {
        long long thr = (long long)N_EDGES * 32;
        sage_scatter<<<(int)((thr + BLK - 1) / BLK), BLK, 0, stream>>>(
            h, src1, dst1, agg, deg, N_EDGES, HID_DIM);
    }
    {
        int total = N_NODES * HID_DIM;
        sage_normalize<<<(total + BLK - 1) / BLK, BLK, 0, stream>>>(
            agg, deg, N_NODES, HID_DIM);
    }
    {
        int waves = (N_NODES / 16) * (OUT_DIM / 16);
        long long thr = (long long)waves * 32;
        sage_gemm_relu<<<(int)((thr + BLK - 1) / BLK), BLK, 0, stream>>>(
            h, agg, W2, b2, out, N_NODES, HID_DIM, OUT_DIM);
    }
}